// GraphAttention_76510547411180
// MI455X (gfx1250) — compile-verified
//
#include <hip/hip_runtime.h>
#include <math.h>

// ---------------- problem constants (from reference) ----------------
#define N_NODES 100000
#define N_EDGES 1600000
#define NODE_F  128
#define EDGE_F  64
#define OUT_F   32
#define NEG_SLOPE 0.01f

typedef __attribute__((ext_vector_type(2))) float v2f;
typedef __attribute__((ext_vector_type(8))) float v8f;

// ---------------- float <-> monotonic signed-int encoding ----------------
// Order-preserving map so we can use native global_atomic_max_i32 for
// the float segment-max (no CAS loop).
__device__ __forceinline__ int enc_f32(float f) {
    int i = __float_as_int(f);
    return i >= 0 ? i : (i ^ 0x7FFFFFFF);
}
__device__ __forceinline__ float dec_f32(int i) {
    return __int_as_float(i >= 0 ? i : (i ^ 0x7FFFFFFF));
}
#define SEG_NEG_INF ((int)0x807FFFFF) // enc(-inf)

// ---------------- workspace layout (float offsets) ----------------
// total ~20.8 MB
static constexpr size_t WS_Z     = 0;                              // z: N_NODES*OUT_F
static constexpr size_t WS_SSRC  = WS_Z + (size_t)N_NODES * OUT_F; // s_src: N_NODES
static constexpr size_t WS_SDST  = WS_SSRC + N_NODES;              // s_dst: N_NODES
static constexpr size_t WS_DENOM = WS_SDST + N_NODES;              // denom: N_NODES
static constexpr size_t WS_SEG   = WS_DENOM + N_NODES;             // seg (int): N_NODES
static constexpr size_t WS_WEA   = WS_SEG + N_NODES;               // w_ea: 64
static constexpr size_t WS_LOGIT = WS_WEA + EDGE_F;                // logits: N_EDGES

// ---------------- kernel 0: init accumulators ----------------
__global__ void gat_init_kernel(float* __restrict__ out,
                                float* __restrict__ denom,
                                int*   __restrict__ seg) {
    int t = blockIdx.x * blockDim.x + threadIdx.x;
    if (t < N_NODES * OUT_F) out[t] = 0.0f;
    if (t < N_NODES) { denom[t] = 0.0f; seg[t] = SEG_NEG_INF; }
}

// ---------------- kernel 1: z = h @ W_node via V_WMMA_F32_16X16X4_F32 ----------------
// One wave32 computes a 16x32 output tile as two 16x16 WMMA accumulators,
// sweeping K=128 in steps of 4. W_node (128x32 f32 = 16 KB) is staged in LDS
// in a WMMA-fragment-ready layout:
//     lw[p*64 + swz(n,p)] = { W[2p][n], W[2p+1][n] }   (pair of k-rows adjacent)
// with a 32-word skew on odd k-pairs so the two 16-lane halves (which read
// k-pairs p and p+1) hit disjoint banks -> one conflict-free ds_load_b64 per
// B fragment, no register repacking.
//
// Fragment layouts (ISA 7.12.2):
//   A 16x4 : lanes 0-15 -> M=lane,    VGPR0=K0, VGPR1=K1
//            lanes16-31 -> M=lane-16, VGPR0=K2, VGPR1=K3
//   B 4x16 : VGPRv, lanes 0-15 -> K=v   (N=lane)
//            VGPRv, lanes16-31 -> K=v+2 (N=lane-16)
//   C 16x16: VGPRi, lanes 0-15 -> (M=i,   N=lane), lanes16-31 -> (M=i+8, N=lane-16)
//
// The C accumulators already hold z in registers, so the per-node attention
// scalars s_src[n]=z[n]·a_src, s_dst[n]=z[n]·a_dst are reduced in-register
// with __shfl_xor (masks 8/4/2/1 stay inside each 16-lane half), saving a
// separate 12.8 MB pass over z.
__global__ void __launch_bounds__(128)
gat_node_gemm_wmma(const float* __restrict__ h,
                   const float* __restrict__ Wn,
                   const float* __restrict__ aw,
                   float* __restrict__ z,
                   float* __restrict__ s_src,
                   float* __restrict__ s_dst) {
    __shared__ float lw[NODE_F * OUT_F]; // 16 KB of the 320 KB WGP LDS
    for (int idx = threadIdx.x; idx < NODE_F * OUT_F; idx += 128) {
        int k = idx >> 5;   // W row 0..127
        int n = idx & 31;   // W col 0..31
        int p = k >> 1;     // k-pair
        int off = ((n << 1) + ((p & 1) << 5)) & 63; // bank skew for odd pairs
        lw[(p << 6) + off + (k & 1)] = Wn[idx];
    }
    __syncthreads();

    const int wave = threadIdx.x >> 5;
    const int lane = threadIdx.x & 31;
    const int tile = blockIdx.x * 4 + wave;     // 16-row tile index
    const int row0 = tile * 16;
    if (row0 >= N_NODES) return;                // wave-uniform: EXEC stays all-1s

    const int half = lane >> 4;                 // 0: lanes 0-15, 1: lanes 16-31
    const int m16  = lane & 15;
    int arow = row0 + m16;
    if (arow >= N_NODES) arow = N_NODES - 1;    // clamp reads, mask stores below
    const float* ap = h + (size_t)arow * NODE_F + half * 2;

    v8f c0 = {}; // output cols 0..15
    v8f c1 = {}; // output cols 16..31
#pragma unroll
    for (int kk = 0; kk < NODE_F; kk += 4) {
        v2f a = *(const v2f*)(ap + kk);         // 8B-aligned global_load_b64
        const int p0 = (kk >> 1) + half;        // this half-wave's k-pair
        const int sk = (p0 & 1) << 5;
        v2f b0 = *(const v2f*)&lw[(p0 << 6) + (((m16       << 1) + sk) & 63)];
        v2f b1 = *(const v2f*)&lw[(p0 << 6) + ((((m16 + 16) << 1) + sk) & 63)];
        c0 = __builtin_amdgcn_wmma_f32_16x16x4_f32(false, a, false, b0,
                                                   (short)0, c0, false, false);
        c1 = __builtin_amdgcn_wmma_f32_16x16x4_f32(false, a, false, b1,
                                                   (short)0, c1, false, false);
    }

    // attention weights for this lane's two output columns
    const float as0 = aw[m16],          as1 = aw[16 + m16];      // a_src
    const float ad0 = aw[OUT_F + m16],  ad1 = aw[OUT_F + 16 + m16]; // a_dst

#pragma unroll
    for (int i = 0; i < 8; ++i) {
        const int rr = row0 + i + half * 8;
        float ts = c0[i] * as0 + c1[i] * as1;
        float td = c0[i] * ad0 + c1[i] * ad1;
#pragma unroll
        for (int m = 8; m >= 1; m >>= 1) {      // reduce within each 16-lane half
            ts += __shfl_xor(ts, m, 32);
            td += __shfl_xor(td, m, 32);
        }
        if (rr < N_NODES) {
            z[(size_t)rr * OUT_F + m16]      = c0[i];
            z[(size_t)rr * OUT_F + 16 + m16] = c1[i];
            if (m16 == 0) { s_src[rr] = ts; s_dst[rr] = td; }
        }
    }
}

// ---------------- kernel 2: w_ea = W_edge @ a_edge (64 floats) ----------------
// Collapses ez@a_edge so the full 1.6Mx32 `ez` matrix is never materialized.
__global__ void gat_edge_weight_kernel(const float* __restrict__ We,
                                       const float* __restrict__ aw,
                                       float* __restrict__ w_ea) {
    int k = threadIdx.x;
    if (k < EDGE_F) {
        float s = 0.0f;
#pragma unroll
        for (int j = 0; j < OUT_F; ++j) s += We[k * OUT_F + j] * aw[2 * OUT_F + j];
        w_ea[k] = s;
    }
}

// ---------------- kernel 3: edge logits + segment max ----------------
// Streams edge_feat once (410 MB, the dominant traffic on this problem).
__global__ void __launch_bounds__(256)
gat_edge_logits_kernel(const float* __restrict__ ef,
                       const int*   __restrict__ src,
                       const int*   __restrict__ dst,
                       const float* __restrict__ w_ea,
                       const float* __restrict__ s_src,
                       const float* __restrict__ s_dst,
                       float* __restrict__ logits,
                       int*   __restrict__ seg) {
    __shared__ float lw[EDGE_F];
    if (threadIdx.x < EDGE_F) lw[threadIdx.x] = w_ea[threadIdx.x];
    __syncthreads();

    int e = blockIdx.x * blockDim.x + threadIdx.x;
    if (e >= N_EDGES) return;
    const float4* ep = (const float4*)(ef + (size_t)e * EDGE_F);
    float dot = 0.0f;
#pragma unroll
    for (int q = 0; q < 16; ++q) {
        float4 v = ep[q];
        dot += v.x * lw[4 * q + 0] + v.y * lw[4 * q + 1]
             + v.z * lw[4 * q + 2] + v.w * lw[4 * q + 3];
    }
    const int s = src[e], d = dst[e];
    float t = dot + s_src[s] + s_dst[d];
    float lr = t > 0.0f ? t : NEG_SLOPE * t; // leaky_relu
    logits[e] = lr;
    atomicMax(&seg[d], enc_f32(lr));         // -> native i32 atomic max in L2
}

// ---------------- kernel 4: exp + segment sums (denominator & numerator) ----------------
// z (12.8 MB) is L2-resident on the 192 MB L2; gathers and f32 atomics stay on-chip.
__global__ void __launch_bounds__(256)
gat_edge_accum_kernel(const float* __restrict__ logits,
                      const int*   __restrict__ src,
                      const int*   __restrict__ dst,
                      const int*   __restrict__ seg,
                      const float* __restrict__ z,
                      float* __restrict__ denom,
                      float* __restrict__ out) {
    int e = blockIdx.x * blockDim.x + threadIdx.x;
    if (e >= N_EDGES) return;
    const int s = src[e], d = dst[e];
    const float m  = dec_f32(seg[d]);        // finite: d has >=1 edge (this one)
    const float ev = expf(logits[e] - m);
    atomicAdd(&denom[d], ev);
    const float4* zp = (const float4*)(z + (size_t)s * OUT_F);
    float* op = out + (size_t)d * OUT_F;
#pragma unroll
    for (int q = 0; q < 8; ++q) {
        float4 v = zp[q];
        atomicAdd(op + 4 * q + 0, ev * v.x);
        atomicAdd(op + 4 * q + 1, ev * v.y);
        atomicAdd(op + 4 * q + 2, ev * v.z);
        atomicAdd(op + 4 * q + 3, ev * v.w);
    }
}

// ---------------- kernel 5: u = num / max(denom, 1e-20) ----------------
__global__ void gat_finalize_kernel(const float* __restrict__ denom,
                                    float* __restrict__ out) {
    int t = blockIdx.x * blockDim.x + threadIdx.x;
    if (t >= N_NODES * OUT_F) return;
    float dn = denom[t >> 5]; // OUT_F == 32
    out[t] = out[t] / fmaxf(dn, 1e-20f);
}

// ---------------- launch ----------------
extern "C" void kernel_launch(void* const* d_in, const int* in_sizes, int n_in,
                              void* d_out, int out_size, void* d_ws, size_t ws_size,
                              hipStream_t stream) {
    const float* h   = (const float*)d_in[0];
    const float* ef  = (const float*)d_in[1];
    const int*   src = (const int*)d_in[2];
    const int*   dst = (const int*)d_in[3];
    const float* Wn  = (const float*)d_in[4];
    const float* We  = (const float*)d_in[5];
    const float* aw  = (const float*)d_in[6];
    float* out = (float*)d_out;

    float* ws      = (float*)d_ws;
    float* z       = ws + WS_Z;
    float* s_src   = ws + WS_SSRC;
    float* s_dst   = ws + WS_SDST;
    float* denom   = ws + WS_DENOM;
    int*   seg     = (int*)(ws + WS_SEG);
    float* w_ea    = ws + WS_WEA;
    float* logits  = ws + WS_LOGIT;

    const int elems = N_NODES * OUT_F;

    gat_init_kernel<<<(elems + 255) / 256, 256, 0, stream>>>(out, denom, seg);

    // 6250 row tiles of 16, 4 waves (tiles) per 128-thread block
    const int ntiles  = (N_NODES + 15) / 16;
    const int nblocks = (ntiles + 3) / 4;
    gat_node_gemm_wmma<<<nblocks, 128, 0, stream>>>(h, Wn, aw, z, s_src, s_dst);

    gat_edge_weight_kernel<<<1, 64, 0, stream>>>(We, aw, w_ea);

    gat_edge_logits_kernel<<<(N_EDGES + 255) / 256, 256, 0, stream>>>(
        ef, src, dst, w_ea, s_src, s_dst, logits, seg);

    gat_edge_accum_kernel<<<(N_EDGES + 255) / 256, 256, 0, stream>>>(
        logits, src, dst, seg, z, denom, out);

    gat_finalize_kernel<<<(elems + 255) / 256, 256, 0, stream>>>(denom, out);
}